// TrainableMultiVectorEncoder_36807869726945
// MI455X (gfx1250) — compile-verified
//
#include <hip/hip_runtime.h>
#include <hip/hip_bf16.h>

// ---------------------------------------------------------------------------
// Problem constants (from reference): B=256, T=261, D=768, NUM_REG=4, G=16,
// ROI=3.  M_total = B*T = 66816 rows.  GEMM: t = tokens @ W^T + b.
// ---------------------------------------------------------------------------
#define BATCH 256
#define TSEQ  261
#define DIM   768
#define NCUE  5            // 1 + NUM_REG
#define NPAT  256          // G*G
#define MTOT  (BATCH * TSEQ)

typedef __attribute__((ext_vector_type(2))) float v2f;
typedef __attribute__((ext_vector_type(8))) float v8f;

// ---------------------------------------------------------------------------
// Kernel 1: fp32 WMMA GEMM   t[M][N] = sum_k tokens[M][k] * W[N][k] + bias[N]
// Block tile 128x64, K staged via LDS in chunks of 32.  8 waves per block,
// each wave owns a 2x2 grid of 16x16 fragments (32x32 output) so every LDS
// fragment read feeds two V_WMMA_F32_16X16X4_F32 ops.
// ---------------------------------------------------------------------------
#define BM 128
#define BN 64
#define BK 32
#define LDS_STRIDE (BK + 2)   // 34 floats: even -> float2 loads stay 8B aligned

__global__ __launch_bounds__(256, 2)
void proj_gemm_wmma(const float* __restrict__ A,      // tokens  [MTOT][DIM]
                    const float* __restrict__ W,      // W_proj  [DIM][DIM]
                    const float* __restrict__ bias,   // b_proj  [DIM]
                    float* __restrict__ T)            // out t   [MTOT][DIM]
{
    __shared__ float As[BM][LDS_STRIDE];
    __shared__ float Bs[BN][LDS_STRIDE];   // Bs[n][k] = W[bn+n][k0+k]

    const int tid  = threadIdx.x;
    const int wave = tid >> 5;
    const int lane = tid & 31;
    const int hl   = lane >> 4;    // half-wave select (ISA A/B/C lane split)
    const int l16  = lane & 15;

    const int bm = blockIdx.x * BM;
    const int bn = blockIdx.y * BN;

    // 8 M-subtiles x 4 N-subtiles; each wave takes 2 adjacent of each.
    const int sm0 = (wave & 3) * 2;
    const int sm1 = sm0 + 1;
    const int sn0 = (wave >> 2) * 2;
    const int sn1 = sn0 + 1;

    v8f c00 = {}, c01 = {}, c10 = {}, c11 = {};

    for (int k0 = 0; k0 < DIM; k0 += BK) {
        // ---- stage A chunk (128x32) into LDS: 1024 float4 slots ----------
        #pragma unroll
        for (int i = 0; i < 4; ++i) {
            const int linear = tid + i * 256;
            const int r  = linear >> 3;             // 8 float4 per 32-col row
            const int cc = (linear & 7) * 4;
            const float* ga = A + (size_t)(bm + r) * DIM + (k0 + cc);
            As[r][cc + 0] = ga[0];
            As[r][cc + 1] = ga[1];
            As[r][cc + 2] = ga[2];
            As[r][cc + 3] = ga[3];
        }
        // ---- stage B chunk (64x32): 512 float4 slots ---------------------
        #pragma unroll
        for (int i = 0; i < 2; ++i) {
            const int linear = tid + i * 256;
            const int r  = linear >> 3;
            const int cc = (linear & 7) * 4;
            const float* gb = W + (size_t)(bn + r) * DIM + (k0 + cc);
            Bs[r][cc + 0] = gb[0];
            Bs[r][cc + 1] = gb[1];
            Bs[r][cc + 2] = gb[2];
            Bs[r][cc + 3] = gb[3];
        }
        // hint next K chunk into cache (global_prefetch_b8)
        if (k0 + BK < DIM) {
            __builtin_prefetch(A + (size_t)(bm + (tid >> 1)) * DIM + k0 + BK, 0, 0);
            __builtin_prefetch(W + (size_t)(bn + (tid & 63)) * DIM + k0 + BK, 0, 0);
        }
        __syncthreads();

        // ---- 8 WMMA K-steps of 4: 2 A frags + 2 B frags -> 4 WMMAs -------
        #pragma unroll
        for (int kk = 0; kk < BK; kk += 4) {
            // A 16x4 frag: lanes0-15 {K=0,1}, lanes16-31 {K=2,3}, M = l16
            v2f a0 = *(const v2f*)&As[sm0 * 16 + l16][kk + 2 * hl];
            v2f a1 = *(const v2f*)&As[sm1 * 16 + l16][kk + 2 * hl];
            // B 4x16 frag: N = l16, same K split per half-wave
            v2f b0 = *(const v2f*)&Bs[sn0 * 16 + l16][kk + 2 * hl];
            v2f b1 = *(const v2f*)&Bs[sn1 * 16 + l16][kk + 2 * hl];
            c00 = __builtin_amdgcn_wmma_f32_16x16x4_f32(
                      false, a0, false, b0, (short)0, c00, false, false);
            c01 = __builtin_amdgcn_wmma_f32_16x16x4_f32(
                      false, a0, false, b1, (short)0, c01, false, false);
            c10 = __builtin_amdgcn_wmma_f32_16x16x4_f32(
                      false, a1, false, b0, (short)0, c10, false, false);
            c11 = __builtin_amdgcn_wmma_f32_16x16x4_f32(
                      false, a1, false, b1, (short)0, c11, false, false);
        }
        __syncthreads();
    }

    // ---- epilogue: bias + store.  C layout: VGPR r -> M = r + 8*hl, N=l16.
    const int row0 = bm + sm0 * 16;
    const int row1 = bm + sm1 * 16;
    const int col0 = bn + sn0 * 16 + l16;
    const int col1 = bn + sn1 * 16 + l16;
    const float bias0 = bias[col0];
    const float bias1 = bias[col1];
    #pragma unroll
    for (int r = 0; r < 8; ++r) {
        const int ra = row0 + r + 8 * hl;
        const int rb = row1 + r + 8 * hl;
        T[(size_t)ra * DIM + col0] = c00[r] + bias0;
        T[(size_t)ra * DIM + col1] = c01[r] + bias1;
        T[(size_t)rb * DIM + col0] = c10[r] + bias0;
        T[(size_t)rb * DIM + col1] = c11[r] + bias1;
    }
}

// ---------------------------------------------------------------------------
// Kernel 2: per (batch, cue) block: sim -> argmax -> 3x3 ROI mean -> L2 norm.
// ---------------------------------------------------------------------------
__global__ __launch_bounds__(256)
void buddy_pool_norm(const float* __restrict__ T, float* __restrict__ out)
{
    const int bc = blockIdx.x;
    const int b  = bc / NCUE;
    const int c  = bc % NCUE;
    const int tid = threadIdx.x;

    __shared__ float cue[DIM];
    __shared__ float roi[DIM];
    __shared__ float redv[256];
    __shared__ int   redi[256];

    const float* tb   = T + (size_t)b * TSEQ * DIM;
    const float* cueg = tb + (size_t)c * DIM;
    const float* pat  = tb + (size_t)NCUE * DIM;   // 256 patch rows

    for (int d = tid; d < DIM; d += 256) cue[d] = cueg[d];
    __syncthreads();

    // similarity: one patch per thread
    float sim = 0.f;
    {
        const float* pr = pat + (size_t)tid * DIM;
        #pragma unroll 4
        for (int d = 0; d < DIM; ++d) sim += cue[d] * pr[d];
    }
    redv[tid] = sim;
    redi[tid] = tid;
    __syncthreads();

    // argmax (first max wins, matching jnp.argmax)
    for (int s = 128; s > 0; s >>= 1) {
        if (tid < s) {
            const float ov = redv[tid + s];
            const int   oi = redi[tid + s];
            if (ov > redv[tid] || (ov == redv[tid] && oi < redi[tid])) {
                redv[tid] = ov;
                redi[tid] = oi;
            }
        }
        __syncthreads();
    }
    const int idx = redi[0];
    const int h = idx >> 4, w = idx & 15;
    const int r0 = max(h - 1, 0), r1 = min(h + 1, 15);
    const int q0 = max(w - 1, 0), q1 = min(w + 1, 15);
    const float inv = 1.0f / (float)((r1 - r0 + 1) * (q1 - q0 + 1));

    // ROI mean
    for (int d = tid; d < DIM; d += 256) {
        float acc = 0.f;
        for (int hh = r0; hh <= r1; ++hh)
            for (int ww = q0; ww <= q1; ++ww)
                acc += pat[(size_t)(hh * 16 + ww) * DIM + d];
        roi[d] = acc * inv;
    }
    __syncthreads();

    // L2 norms of cue and roi
    float s0 = 0.f, s1 = 0.f;
    for (int d = tid; d < DIM; d += 256) {
        s0 += cue[d] * cue[d];
        s1 += roi[d] * roi[d];
    }
    redv[tid] = s0;
    __syncthreads();
    for (int s = 128; s > 0; s >>= 1) {
        if (tid < s) redv[tid] += redv[tid + s];
        __syncthreads();
    }
    const float ncue = sqrtf(redv[0]);
    __syncthreads();
    redv[tid] = s1;
    __syncthreads();
    for (int s = 128; s > 0; s >>= 1) {
        if (tid < s) redv[tid] += redv[tid + s];
        __syncthreads();
    }
    const float nroi = sqrtf(redv[0]);

    const float icue = 1.0f / fmaxf(ncue, 1e-12f);
    const float iroi = 1.0f / fmaxf(nroi, 1e-12f);

    float* ob = out + (size_t)b * (2 * NCUE) * DIM;
    for (int d = tid; d < DIM; d += 256) {
        ob[(size_t)c * DIM + d]          = cue[d] * icue;
        ob[(size_t)(NCUE + c) * DIM + d] = roi[d] * iroi;
    }
}

// ---------------------------------------------------------------------------
extern "C" void kernel_launch(void* const* d_in, const int* in_sizes, int n_in,
                              void* d_out, int out_size, void* d_ws, size_t ws_size,
                              hipStream_t stream)
{
    const float* tokens = (const float*)d_in[0];   // [B,T,D] f32
    const float* Wproj  = (const float*)d_in[1];   // [D,D]   f32
    const float* bproj  = (const float*)d_in[2];   // [D]     f32
    float* out = (float*)d_out;                    // [B,10,D] f32
    float* t   = (float*)d_ws;                     // [MTOT,D] f32 scratch

    dim3 g1(MTOT / BM, DIM / BN);   // 522 x 12 blocks
    proj_gemm_wmma<<<g1, 256, 0, stream>>>(tokens, Wproj, bproj, t);

    buddy_pool_norm<<<BATCH * NCUE, 256, 0, stream>>>(t, out);
}